// MultiHeadDistance_81990925680694
// MI455X (gfx1250) — compile-verified
//
#include <hip/hip_runtime.h>

// ---------------------------------------------------------------------------
// MultiHeadDistance on gfx1250 (MI455X):
//   out[n,m] = max_h  ( s[n,h,:]/|s[n,h,:]| ) . ( ro[m,h,:]/|ro[m,h,:]| )
// N=2048, M=8192, H=8, D=32.  FP32 throughout via V_WMMA_F32_16X16X4_F32.
//
// v2: LDS-staged fragments. Global loads are fully coalesced b128; WMMA
// fragments come from LDS with a 36-dword padded row stride that makes every
// ds_load_b64 bank-conflict-free (36r+c mod 64 covers all 64 banks for
// r in 0..15, c in 0..3). Staging is wave-private -> zero barriers.
// ---------------------------------------------------------------------------

typedef __attribute__((ext_vector_type(2))) float v2f;
typedef __attribute__((ext_vector_type(8))) float v8f;

#define HEADS 8
#define HDIM  32
#define ROWF  256   // floats per input row (HEADS * HDIM)
#define PADW  36    // padded LDS row stride in dwords (32 data + 4 pad)
#define LDS_A (16 * PADW)              // 576 dwords per wave
#define LDS_B (64 * PADW)              // 2304 dwords per wave
#define LDS_W (LDS_A + LDS_B)          // 2880 dwords per wave
#define WAVES 4                        // waves per block

// ---------------------------------------------------------------------------
// Kernel 1: reciprocal L2 norm of every (row, head) 32-float vector.
// One wave32 per vector; butterfly reduction across the 32 lanes.
// ---------------------------------------------------------------------------
__global__ void mhd_rnorm(const float* __restrict__ in,
                          float* __restrict__ rnorm, int nvec) {
    int gid  = blockIdx.x * blockDim.x + threadIdx.x;
    int vec  = gid >> 5;
    int lane = threadIdx.x & 31;
    if (vec >= nvec) return;
    float x  = in[(size_t)vec * HDIM + lane];
    float ss = x * x;
#pragma unroll
    for (int m = 16; m >= 1; m >>= 1)
        ss += __shfl_xor(ss, m, 32);
    if (lane == 0)
        rnorm[vec] = 1.0f / sqrtf(ss);
}

// ---------------------------------------------------------------------------
// Kernel 2: fused 8-head GEMM + max-over-heads.
//   Wave tile: 16 rows (n) x 64 cols (m) -> 4 accumulators sharing 1 A frag.
//   Block: 4 waves = 16 n-rows x 256 m-cols.
//
// WMMA f32 16x16x4 fragment layouts (ISA 7.12.2, 32-bit):
//   A (16x4): lane<16 -> row=lane,    {K0,K1}; lane>=16 -> row=lane-16, {K2,K3}
//   B (4x16): lane<16 -> col=lane,    {K0,K1}; lane>=16 -> {K2,K3}
//   C/D     : VGPR r -> row = r + 8*(lane/16), col = lane%16
// ---------------------------------------------------------------------------
__global__ void __launch_bounds__(128)
mhd_gemm_max(const float* __restrict__ A,      // batch_s  [N][256]
             const float* __restrict__ B,      // batch_ro [M][256]
             const float* __restrict__ rnA,    // [N*8] 1/|s|
             const float* __restrict__ rnB,    // [M*8] 1/|ro|
             float* __restrict__ out,          // [N][M]
             int N, int M) {
    extern __shared__ float smem[];
    const int lane = threadIdx.x & 31;
    const int wave = threadIdx.x >> 5;
    const int lr   = lane & 15;     // row (A) / col (B,D) within tile
    const int hi   = lane >> 4;     // selects K pair {0,1} vs {2,3}

    float* lA = smem + wave * LDS_W;   // wave-private A slab (16 x PADW)
    float* lB = lA + LDS_A;            // wave-private B slab (64 x PADW)

    const int blocksPerRow = M >> 8;                 // M / 256
    const int n0 = (blockIdx.x / blocksPerRow) << 4;
    const int m0 = ((blockIdx.x % blocksPerRow) << 8) + (wave << 6);

    // staging assignment: lane -> (row sr+4i, 4-float chunk sc)
    const int sr = lane >> 3;          // 0..3
    const int sc = (lane & 7) << 2;    // 0,4,...,28

    v8f vmax0, vmax1, vmax2, vmax3;
#pragma unroll
    for (int i = 0; i < 8; ++i) {
        vmax0[i] = -3.0e38f; vmax1[i] = -3.0e38f;
        vmax2[i] = -3.0e38f; vmax3[i] = -3.0e38f;
    }

#pragma unroll
    for (int h = 0; h < HEADS; ++h) {
        // ---- stage this head's A (16x32) and B (64x32) slabs into LDS ----
        // global side: 8 consecutive lanes read 128B contiguous per row.
#pragma unroll
        for (int i = 0; i < 4; ++i) {
            const int r = sr + 4 * i;
            const float4 v = *(const float4*)(A + (size_t)(n0 + r) * ROWF
                                                + h * HDIM + sc);
            *(float4*)(lA + r * PADW + sc) = v;
        }
#pragma unroll
        for (int i = 0; i < 16; ++i) {
            const int r = sr + 4 * i;
            const float4 v = *(const float4*)(B + (size_t)(m0 + r) * ROWF
                                                + h * HDIM + sc);
            *(float4*)(lB + r * PADW + sc) = v;
        }

        // per-lane normalization scales: A row = n0+lr, B rows = m0+16t+lr
        const float sA  = rnA[(size_t)(n0 + lr) * HEADS + h];
        const float sB0 = rnB[(size_t)(m0 +  0 + lr) * HEADS + h];
        const float sB1 = rnB[(size_t)(m0 + 16 + lr) * HEADS + h];
        const float sB2 = rnB[(size_t)(m0 + 32 + lr) * HEADS + h];
        const float sB3 = rnB[(size_t)(m0 + 48 + lr) * HEADS + h];

        v8f c0 = {}, c1 = {}, c2 = {}, c3 = {};
#pragma unroll
        for (int k0 = 0; k0 < HDIM; k0 += 4) {
            const int ka = k0 + 2 * hi;
            v2f a = *(const v2f*)(lA + lr * PADW + ka);
            a.x *= sA; a.y *= sA;                       // normalize A on load
            v2f b0 = *(const v2f*)(lB + (lr +  0) * PADW + ka);
            v2f b1 = *(const v2f*)(lB + (lr + 16) * PADW + ka);
            v2f b2 = *(const v2f*)(lB + (lr + 32) * PADW + ka);
            v2f b3 = *(const v2f*)(lB + (lr + 48) * PADW + ka);
            c0 = __builtin_amdgcn_wmma_f32_16x16x4_f32(false, a, false, b0,
                                                       (short)0, c0, false, false);
            c1 = __builtin_amdgcn_wmma_f32_16x16x4_f32(false, a, false, b1,
                                                       (short)0, c1, false, false);
            c2 = __builtin_amdgcn_wmma_f32_16x16x4_f32(false, a, false, b2,
                                                       (short)0, c2, false, false);
            c3 = __builtin_amdgcn_wmma_f32_16x16x4_f32(false, a, false, b3,
                                                       (short)0, c3, false, false);
        }
        // B-side normalization: column of every c element == lr for this lane,
        // so sB* are uniform per lane across all 8 accumulator VGPRs.
#pragma unroll
        for (int i = 0; i < 8; ++i) {
            vmax0[i] = fmaxf(vmax0[i], c0[i] * sB0);
            vmax1[i] = fmaxf(vmax1[i], c1[i] * sB1);
            vmax2[i] = fmaxf(vmax2[i], c2[i] * sB2);
            vmax3[i] = fmaxf(vmax3[i], c3[i] * sB3);
        }
    }

    // D layout: VGPR r -> out row n0 + r + 8*hi, col m0 + 16t + lr
#pragma unroll
    for (int r = 0; r < 8; ++r) {
        size_t row = (size_t)(n0 + r + 8 * hi) * (size_t)M + (size_t)m0 + lr;
        out[row +  0] = vmax0[r];
        out[row + 16] = vmax1[r];
        out[row + 32] = vmax2[r];
        out[row + 48] = vmax3[r];
    }
}

// ---------------------------------------------------------------------------
extern "C" void kernel_launch(void* const* d_in, const int* in_sizes, int n_in,
                              void* d_out, int out_size, void* d_ws, size_t ws_size,
                              hipStream_t stream) {
    const float* batch_s  = (const float*)d_in[0];
    const float* batch_ro = (const float*)d_in[1];
    const int N = in_sizes[0] / ROWF;   // 2048
    const int M = in_sizes[1] / ROWF;   // 8192

    // workspace: reciprocal norms only ((N+M)*8 floats = 320 KB)
    float* rnA = (float*)d_ws;
    float* rnB = rnA + (size_t)N * HEADS;

    const int nvecA = N * HEADS;
    const int nvecB = M * HEADS;
    mhd_rnorm<<<(nvecA * 32 + 255) / 256, 256, 0, stream>>>(batch_s, rnA, nvecA);
    mhd_rnorm<<<(nvecB * 32 + 255) / 256, 256, 0, stream>>>(batch_ro, rnB, nvecB);

    const int nblocks = (N / 16) * (M / 256);            // 128 * 32 = 4096
    const size_t ldsBytes = (size_t)WAVES * LDS_W * 4;   // 46080 B
    mhd_gemm_max<<<nblocks, WAVES * 32, ldsBytes, stream>>>(
        batch_s, batch_ro, rnA, rnB, (float*)d_out, N, M);
}